// MomGraphConv_36962488549736
// MI455X (gfx1250) — compile-verified
//
#include <hip/hip_runtime.h>

#define NNODES  100000
#define IN_F    256
#define NEDGES  3200000
#define GAMMA   0.1f
#define KCHUNK  64            // B rows staged per TDM transfer (64 KB of LDS)

typedef __attribute__((ext_vector_type(2))) float    v2f;
typedef __attribute__((ext_vector_type(8))) float    v8f;
typedef __attribute__((ext_vector_type(4))) unsigned u32x4;
typedef __attribute__((ext_vector_type(8))) int      i32x8;
typedef __attribute__((ext_vector_type(4))) int      i32x4;

// ---------------------------------------------------------------------------
// Kernel 1: fold the 4-block momentum recurrence into one 256x256 matrix,
// stored QUAD-INTERLEAVED along K:  (r,c) -> Bc[(r>>2)*1024 + c*4 + (r&3)].
//   x = input @ (Bc + I) + g^4 * velocity0,
//   Bc = (1-g)(g^3 B0 + g^2 B1 + g B2 + B3)
// The interleaving makes each lane's WMMA B-fragment (K, K+1 at column n) an
// adjacent 8-byte pair in LDS -> one conflict-free ds_load_b64 per fragment.
// ---------------------------------------------------------------------------
__global__ __launch_bounds__(256)
void mgc_combine(const float* __restrict__ blocks, float* __restrict__ Bc) {
  int i = blockIdx.x * 256 + threadIdx.x;           // 0 .. 65535
  const int M = IN_F * IN_F;
  float b0 = blocks[i];
  float b1 = blocks[M + i];
  float b2 = blocks[2 * M + i];
  float b3 = blocks[3 * M + i];
  float w = (1.0f - GAMMA) *
            (GAMMA * GAMMA * GAMMA * b0 + GAMMA * GAMMA * b1 + GAMMA * b2 + b3);
  int r = i >> 8, c = i & 255;
  if (r == c) w += 1.0f;                            // + identity (x = v + input)
  Bc[(r >> 2) * (4 * IN_F) + c * 4 + (r & 3)] = w;  // quad-interleaved store
}

// ---------------------------------------------------------------------------
// Kernel 2: init out[n,f] = bias[f]
// ---------------------------------------------------------------------------
__global__ __launch_bounds__(256)
void mgc_init_out(const float* __restrict__ bias, float* __restrict__ out) {
  int i = blockIdx.x * 256 + threadIdx.x;           // float4 index over out
  float4 b = ((const float4*)bias)[i & 63];         // 64 float4 per 256-f row
  ((float4*)out)[i] = b;
}

// ---------------------------------------------------------------------------
// Kernel 3: X = input @ Bc' + g^4 * velocity0.
// One wave owns a 16-row slab x full N=256 (16 accumulator tiles, 128 VGPRs).
// B (quad-interleaved) is DMA'd into LDS by the Tensor Data Mover in 64-row
// chunks; each k-step: one global_load_b64 of A feeds 16 WMMAs whose B
// fragments are single conflict-free ds_load_b64s.
// ---------------------------------------------------------------------------
__global__ __launch_bounds__(256)
void mgc_gemm(const float* __restrict__ A, const float* __restrict__ V0,
              const float* __restrict__ B, float* __restrict__ X, int nrows) {
  __shared__ float Bs[KCHUNK * IN_F];               // 64 KB staging tile

  int wave = threadIdx.x >> 5;
  int lane = threadIdx.x & 31;
  int m0 = (blockIdx.x * 8 + wave) * 16;            // wave-uniform
  bool active = (m0 < nrows);

  int mrow  = lane & 15;                            // M (or N) index in tile
  int khalf = (lane >> 4) << 1;                     // lanes 0-15: K{0,1}; 16-31: K{2,3}
  const float* arow = A + (size_t)(m0 + mrow) * IN_F;

  v8f zacc = {};
  v8f acc[16];
#pragma unroll
  for (int nt = 0; nt < 16; ++nt) acc[nt] = zacc;

  unsigned lds_base = (unsigned)(uintptr_t)(void*)&Bs[0];

  for (int kb = 0; kb < IN_F / KCHUNK; ++kb) {
    if (wave == 0) {
      // ---- TDM: async DMA of a contiguous 64-row (64 KB) chunk of B ----
      unsigned long long ga =
          (unsigned long long)(uintptr_t)(B + (size_t)kb * KCHUNK * IN_F);
      u32x4 g0;
      g0[0] = 1u;                                   // count=1 (valid user D#)
      g0[1] = lds_base;                             // lds_addr
      g0[2] = (unsigned)(ga & 0xffffffffu);         // global_addr[31:0]
      g0[3] = (unsigned)((ga >> 32) & 0x1ffffffu)   // global_addr[56:32]
              | (2u << 30);                         // type = 2 ("image")
      i32x8 g1;
      g1[0] = 0x00020000;                           // data_size=2 -> 4 bytes
      g1[1] = (int)((IN_F & 0xffff) << 16);         // tensor_dim0[15:0]
      g1[2] = (int)((IN_F & 0xffff) << 16);         // tensor_dim0 hi=0 | tensor_dim1 lo
      g1[3] = (int)((IN_F & 0xffff) << 16);         // tensor_dim1 hi=0 | tile_dim0=256
      g1[4] = KCHUNK;                               // tile_dim1=64, tile_dim2=0
      g1[5] = IN_F;                                 // tensor_dim0_stride = 256
      g1[6] = 0;                                    // stride0 hi | stride1 lo (2D)
      g1[7] = 0;                                    // stride1 hi (unused)
      i32x4 gz4 = {0, 0, 0, 0};
      i32x8 gz8 = {0, 0, 0, 0, 0, 0, 0, 0};
      __builtin_amdgcn_tensor_load_to_lds(g0, g1, gz4, gz4, gz8, 0);
      __builtin_amdgcn_s_wait_tensorcnt(0);
    }
    __syncthreads();                                // tile visible to all waves

    if (active) {
      const float* ak = arow + kb * KCHUNK;
      for (int k = 0; k < KCHUNK; k += 4) {
        v2f a;
        a.x = ak[k + khalf];
        a.y = ak[k + khalf + 1];
        // quad-interleaved: rows (k..k+3) of this chunk live at
        // Bs[(k>>2)*1024 + n*4 + (row&3)]; lane wants (k+khalf, k+khalf+1) @ n
        const v2f* bq = (const v2f*)&Bs[(k >> 2) * (4 * IN_F) + mrow * 4 + khalf];
#pragma unroll
        for (int nt = 0; nt < 16; ++nt) {
          v2f b = bq[nt * 32];                      // +n*4 floats per 16 cols
          acc[nt] = __builtin_amdgcn_wmma_f32_16x16x4_f32(
              false, a, false, b, (short)0, acc[nt], false, false);
        }
      }
    }
    __syncthreads();                                // drain readers before overwrite
  }

  if (active) {
    // D layout: VGPR r -> (M=r, N=lane) lanes 0-15; (M=r+8, N=lane-16) lanes 16-31
    int mbase = m0 + ((lane >> 4) << 3);
#pragma unroll
    for (int nt = 0; nt < 16; ++nt) {
      int nn = nt * 16 + mrow;
#pragma unroll
      for (int r = 0; r < 8; ++r) {
        size_t idx = (size_t)(mbase + r) * IN_F + nn;
        X[idx] = acc[nt][r] + 1.0e-4f * V0[idx];    // + g^4 * velocity0
      }
    }
  }
}

// ---------------------------------------------------------------------------
// Kernel 4: SpMM scatter: out[row] += w * X[col]. X (102 MB) is L2-resident.
// One wave per edge: 2 x float4 gather per lane, 8 f32 global atomics.
// ---------------------------------------------------------------------------
__global__ __launch_bounds__(256)
void mgc_spmm(const int* __restrict__ erow, const int* __restrict__ ecol,
              const float* __restrict__ eval, const float* __restrict__ X,
              float* __restrict__ out, int nedges) {
  int e    = blockIdx.x * 8 + (threadIdx.x >> 5);   // wave-uniform edge id
  int lane = threadIdx.x & 31;
  if (e >= nedges) return;

  int   r = erow[e];
  int   c = ecol[e];
  float w = eval[e];

  const float4* xr = (const float4*)(X + (size_t)c * IN_F);
  float* orow = out + (size_t)r * IN_F;

#pragma unroll
  for (int half = 0; half < 2; ++half) {
    int q = lane + half * 32;                       // float4 index in row
    float4 v = xr[q];
    int base = q * 4;
    unsafeAtomicAdd(orow + base + 0, w * v.x);
    unsafeAtomicAdd(orow + base + 1, w * v.y);
    unsafeAtomicAdd(orow + base + 2, w * v.z);
    unsafeAtomicAdd(orow + base + 3, w * v.w);
  }
}

// ---------------------------------------------------------------------------
extern "C" void kernel_launch(void* const* d_in, const int* in_sizes, int n_in,
                              void* d_out, int out_size, void* d_ws, size_t ws_size,
                              hipStream_t stream) {
  const float* input  = (const float*)d_in[0];
  const float* vel0   = (const float*)d_in[1];
  const float* blocks = (const float*)d_in[2];
  const float* bias   = (const float*)d_in[3];
  const int*   erow   = (const int*)d_in[4];
  const int*   ecol   = (const int*)d_in[5];
  const float* evalp  = (const float*)d_in[6];
  float* out = (float*)d_out;

  float* X  = (float*)d_ws;                         // NNODES*IN_F f32 (102.4 MB)
  float* Bc = X + (size_t)NNODES * IN_F;            // IN_F*IN_F f32 (interleaved)

  mgc_combine <<<(IN_F * IN_F) / 256, 256, 0, stream>>>(blocks, Bc);
  mgc_init_out<<<(NNODES * IN_F / 4) / 256, 256, 0, stream>>>(bias, out);

  int mtiles = NNODES / 16;                         // 6250, exact
  mgc_gemm<<<(mtiles + 7) / 8, 256, 0, stream>>>(input, vel0, Bc, X, NNODES);

  mgc_spmm<<<NEDGES / 8, 256, 0, stream>>>(erow, ecol, evalp, X, out, NEDGES);
}